// GNN_35072702939528
// MI455X (gfx1250) — compile-verified
//
#include <hip/hip_runtime.h>
#include <hip/hip_bf16.h>

#define N_NODES   50000
#define N_EDGES   1600000
#define D_H       128
#define N_GRAPHS  512
#define N_CLASS   10
#define LRELU_A   0.05f
#define BN_EPS    1e-5f

typedef __attribute__((ext_vector_type(2))) float v2f;
typedef __attribute__((ext_vector_type(8))) float v8f;

__device__ __forceinline__ float lk(float v) { return v > 0.f ? v : LRELU_A * v; }

// ---------------------------------------------------------------------------
// z[n][o] = sum_k maybe_leaky(X[n][k]) * W[o][k]
// One wave computes one 16x16 tile via V_WMMA_F32_16X16X4_F32 over K=128.
// A (16x4 f32): lane m=lane&15 holds {A[m][k0+2h], A[m][k0+2h+1]}, h=lane>>4.
// B (4x16 f32): lane n=lane&15 holds {B[k0+2h][n], B[k0+2h+1][n]} = W[n][k..].
// ---------------------------------------------------------------------------
__global__ __launch_bounds__(256) void gat_gemm_wmma(const float* __restrict__ X,
                                                     const float* __restrict__ W,
                                                     float* __restrict__ Z,
                                                     int apply_leaky)
{
    const int wave = blockIdx.x * (blockDim.x >> 5) + (threadIdx.x >> 5);
    const int lane = threadIdx.x & 31;
    const int row_base = (wave >> 3) << 4;   // node tile (3125 tiles)
    const int col_base = (wave & 7) << 4;    // output-feature tile (8 tiles)
    if (row_base >= N_NODES) return;

    const int m    = lane & 15;
    const int half = lane >> 4;

    const float* xrow = X + (size_t)(row_base + m) * D_H;
    const float* wrow = W + (size_t)(col_base + m) * D_H;

    v8f c = {};
    #pragma unroll
    for (int k0 = 0; k0 < D_H; k0 += 4) {
        const int kk = k0 + (half << 1);
        v2f a = *(const v2f*)(xrow + kk);
        v2f b = *(const v2f*)(wrow + kk);
        if (apply_leaky) { a.x = lk(a.x); a.y = lk(a.y); }
        c = __builtin_amdgcn_wmma_f32_16x16x4_f32(false, a, false, b,
                                                  (short)0, c, false, false);
    }

    // C/D layout: VGPR i, lanes 0-15 -> M=i, lanes 16-31 -> M=i+8; N = lane&15
    const int col = col_base + m;
    #pragma unroll
    for (int i = 0; i < 8; ++i) {
        const int row = row_base + i + (half << 3);
        Z[(size_t)row * D_H + col] = c[i];
    }
}

// ---------------------------------------------------------------------------
// zl[n] = z[n] . a[0:128]   zr[n] = z[n] . a[128:256]   (wave per node)
// ---------------------------------------------------------------------------
__global__ __launch_bounds__(256) void gat_scores(const float* __restrict__ Z,
                                                  const float* __restrict__ a,
                                                  float* __restrict__ zl,
                                                  float* __restrict__ zr)
{
    const int node = blockIdx.x * (blockDim.x >> 5) + (threadIdx.x >> 5);
    const int lane = threadIdx.x & 31;
    if (node >= N_NODES) return;
    const float4 zv = ((const float4*)(Z + (size_t)node * D_H))[lane];
    const float4 al = ((const float4*)a)[lane];
    const float4 ar = ((const float4*)(a + D_H))[lane];
    float sl = zv.x*al.x + zv.y*al.y + zv.z*al.z + zv.w*al.w;
    float sr = zv.x*ar.x + zv.y*ar.y + zv.z*ar.z + zv.w*ar.w;
    #pragma unroll
    for (int off = 16; off; off >>= 1) {
        sl += __shfl_xor(sl, off);
        sr += __shfl_xor(sr, off);
    }
    if (lane == 0) { zl[node] = sl; zr[node] = sr; }
}

// ---------------------------------------------------------------------------
// Per edge: h = exp(leaky(zl[src] + zr[dst])); h_sum[src] += h (atomic, L2)
// ---------------------------------------------------------------------------
__global__ __launch_bounds__(256) void gat_edge_attn(const float* __restrict__ zl,
                                                     const float* __restrict__ zr,
                                                     const int* __restrict__ src,
                                                     const int* __restrict__ dst,
                                                     float* __restrict__ h_edge,
                                                     float* __restrict__ h_sum)
{
    const int e = blockIdx.x * blockDim.x + threadIdx.x;
    if (e >= N_EDGES) return;
    const int s = src[e], d = dst[e];
    const float h = expf(lk(zl[s] + zr[d]));
    h_edge[e] = h;
    atomicAdd(&h_sum[s], h);
}

// ---------------------------------------------------------------------------
// Per edge (wave): out[src] += (h/h_sum[src]) * z[dst]; 4 floats per lane.
// Both z (25.6MB) and out (25.6MB) are L2-resident on MI455X (192MB L2).
// ---------------------------------------------------------------------------
__global__ __launch_bounds__(256) void gat_edge_scatter(const float* __restrict__ h_edge,
                                                        const float* __restrict__ h_sum,
                                                        const int* __restrict__ src,
                                                        const int* __restrict__ dst,
                                                        const float* __restrict__ Z,
                                                        float* __restrict__ out)
{
    const int e    = blockIdx.x * (blockDim.x >> 5) + (threadIdx.x >> 5);
    const int lane = threadIdx.x & 31;
    if (e >= N_EDGES) return;
    const int s = src[e], d = dst[e];
    const float alpha = h_edge[e] / h_sum[s];
    const float4 v = ((const float4*)(Z + (size_t)d * D_H))[lane];
    float* o = out + (size_t)s * D_H + lane * 4;
    atomicAdd(o + 0, alpha * v.x);
    atomicAdd(o + 1, alpha * v.y);
    atomicAdd(o + 2, alpha * v.z);
    atomicAdd(o + 3, alpha * v.w);
}

// ---------------------------------------------------------------------------
// pooled[idx[n]] += leaky(H[n])   (thread handles one float4)
// ---------------------------------------------------------------------------
__global__ __launch_bounds__(256) void gat_pool(const float* __restrict__ H,
                                                const int* __restrict__ idx,
                                                float* __restrict__ pooled)
{
    const int gid = blockIdx.x * blockDim.x + threadIdx.x;
    if (gid >= N_NODES * 32) return;
    const int node = gid >> 5;
    const int q    = gid & 31;
    const float4 v = ((const float4*)(H + (size_t)node * D_H))[q];
    const int g = idx[node];
    float* p = pooled + (size_t)g * D_H + q * 4;
    atomicAdd(p + 0, lk(v.x));
    atomicAdd(p + 1, lk(v.y));
    atomicAdd(p + 2, lk(v.z));
    atomicAdd(p + 3, lk(v.w));
}

// ---------------------------------------------------------------------------
// Per-feature mean / rstd over the 512 graphs (one block, 128 threads)
// ---------------------------------------------------------------------------
__global__ void bn_stats(const float* __restrict__ pooled,
                         float* __restrict__ mu, float* __restrict__ rstd)
{
    const int f = threadIdx.x;
    float s = 0.f, s2 = 0.f;
    for (int g = 0; g < N_GRAPHS; ++g) {
        const float p = pooled[(size_t)g * D_H + f];
        s += p; s2 += p * p;
    }
    const float m   = s * (1.0f / N_GRAPHS);
    const float var = s2 * (1.0f / N_GRAPHS) - m * m;
    mu[f]   = m;
    rstd[f] = rsqrtf(var + BN_EPS);
}

// ---------------------------------------------------------------------------
// Wave per graph: BN-normalize, FC (10 classes), log_softmax.
// ---------------------------------------------------------------------------
__global__ __launch_bounds__(256) void gat_head(const float* __restrict__ pooled,
                                                const float* __restrict__ mu,
                                                const float* __restrict__ rstd,
                                                const float* __restrict__ gamma,
                                                const float* __restrict__ beta,
                                                const float* __restrict__ fcw,
                                                const float* __restrict__ fcb,
                                                float* __restrict__ out)
{
    const int g    = blockIdx.x * (blockDim.x >> 5) + (threadIdx.x >> 5);
    const int lane = threadIdx.x & 31;
    if (g >= N_GRAPHS) return;
    const int f = lane * 4;
    const float4 pv = ((const float4*)(pooled + (size_t)g * D_H))[lane];
    const float n0 = (pv.x - mu[f+0]) * rstd[f+0] * gamma[f+0] + beta[f+0];
    const float n1 = (pv.y - mu[f+1]) * rstd[f+1] * gamma[f+1] + beta[f+1];
    const float n2 = (pv.z - mu[f+2]) * rstd[f+2] * gamma[f+2] + beta[f+2];
    const float n3 = (pv.w - mu[f+3]) * rstd[f+3] * gamma[f+3] + beta[f+3];

    float logits[N_CLASS];
    #pragma unroll
    for (int c = 0; c < N_CLASS; ++c) {
        const float4 w = ((const float4*)(fcw + (size_t)c * D_H))[lane];
        float part = n0*w.x + n1*w.y + n2*w.z + n3*w.w;
        #pragma unroll
        for (int off = 16; off; off >>= 1) part += __shfl_xor(part, off);
        logits[c] = part + fcb[c];
    }
    float mx = logits[0];
    #pragma unroll
    for (int c = 1; c < N_CLASS; ++c) mx = fmaxf(mx, logits[c]);
    float se = 0.f;
    #pragma unroll
    for (int c = 0; c < N_CLASS; ++c) se += expf(logits[c] - mx);
    const float lse = mx + logf(se);
    if (lane < N_CLASS) out[(size_t)g * N_CLASS + lane] = logits[lane] - lse;
}

// ---------------------------------------------------------------------------
extern "C" void kernel_launch(void* const* d_in, const int* in_sizes, int n_in,
                              void* d_out, int out_size, void* d_ws, size_t ws_size,
                              hipStream_t stream)
{
    const float* x     = (const float*)d_in[0];
    const int*   ei    = (const int*)  d_in[1];
    const int*   idx   = (const int*)  d_in[2];
    const float* W1    = (const float*)d_in[3];
    const float* a1    = (const float*)d_in[4];
    const float* W2    = (const float*)d_in[5];
    const float* a2    = (const float*)d_in[6];
    const float* W3    = (const float*)d_in[7];
    const float* a3    = (const float*)d_in[8];
    const float* fcw   = (const float*)d_in[9];
    const float* fcb   = (const float*)d_in[10];
    const float* gamma = (const float*)d_in[11];
    const float* beta  = (const float*)d_in[12];
    const int* src = ei;
    const int* dst = ei + N_EDGES;
    float* outp = (float*)d_out;
    (void)in_sizes; (void)n_in; (void)out_size; (void)ws_size;

    char* ws = (char*)d_ws;
    size_t off = 0;
    auto carve = [&](size_t bytes) -> float* {
        float* p = (float*)(ws + off);
        off += (bytes + 255) & ~(size_t)255;
        return p;
    };
    float* bufZ   = carve((size_t)N_NODES * D_H * 4);   // 25.6 MB
    float* bufH   = carve((size_t)N_NODES * D_H * 4);   // 25.6 MB
    float* zl     = carve((size_t)N_NODES * 4);
    float* zr     = carve((size_t)N_NODES * 4);
    float* hsum   = carve((size_t)N_NODES * 4);
    float* hedge  = carve((size_t)N_EDGES * 4);         // 6.4 MB
    float* pooled = carve((size_t)N_GRAPHS * D_H * 4);
    float* mu     = carve(D_H * 4);
    float* rstd   = carve(D_H * 4);

    const dim3 blk(256);
    const int gemm_blocks    = (N_NODES / 16) * 8 / 8;        // 3125
    const int scores_blocks  = (N_NODES + 7) / 8;             // 6250
    const int edge1_blocks   = (N_EDGES + 255) / 256;         // 6250
    const int edge2_blocks   = (N_EDGES + 7) / 8;             // 200000
    const int pool_blocks    = (N_NODES * 32 + 255) / 256;    // 6250
    const int head_blocks    = (N_GRAPHS + 7) / 8;            // 64

    auto layer = [&](const float* X, const float* W, const float* a, int leaky_in) {
        gat_gemm_wmma<<<gemm_blocks, blk, 0, stream>>>(X, W, bufZ, leaky_in);
        gat_scores<<<scores_blocks, blk, 0, stream>>>(bufZ, a, zl, zr);
        hipMemsetAsync(hsum, 0, (size_t)N_NODES * 4, stream);
        gat_edge_attn<<<edge1_blocks, blk, 0, stream>>>(zl, zr, src, dst, hedge, hsum);
        // bufH (layer input) already fully consumed by the GEMM above;
        // stream ordering makes this reuse safe.
        hipMemsetAsync(bufH, 0, (size_t)N_NODES * D_H * 4, stream);
        gat_edge_scatter<<<edge2_blocks, blk, 0, stream>>>(hedge, hsum, src, dst, bufZ, bufH);
    };

    layer(x,    W1, a1, 0);
    layer(bufH, W2, a2, 1);
    layer(bufH, W3, a3, 1);

    hipMemsetAsync(pooled, 0, (size_t)N_GRAPHS * D_H * 4, stream);
    gat_pool<<<pool_blocks, blk, 0, stream>>>(bufH, idx, pooled);
    bn_stats<<<1, D_H, 0, stream>>>(pooled, mu, rstd);
    gat_head<<<head_blocks, blk, 0, stream>>>(pooled, mu, rstd, gamma, beta, fcw, fcb, outp);
}